// ARPS_69595650064692
// MI455X (gfx1250) — compile-verified
//
#include <hip/hip_runtime.h>
#include <stdint.h>

#define HGT 30
#define WID 30
#define CELLS 900           // floats per grid
#define QCELLS 225          // 16-byte chunks per grid
#define PADF 912            // padded floats per wave region (16B multiple)
#define WAVES 8             // waves (grids) per block

__device__ __forceinline__ int clip30(int x) { return min(max(x, 0), HGT - 1); }

__global__ __launch_bounds__(256) void arc_paint_kernel(
    const float* __restrict__ grids, const int* __restrict__ anchors,
    const int* __restrict__ p_seed_dr, const int* __restrict__ p_seed_dc,
    const int* __restrict__ p_off_r, const int* __restrict__ p_off_c,
    const int* __restrict__ p_color, float* __restrict__ out, int B)
{
    __shared__ __attribute__((aligned(16))) float    cells[WAVES][PADF];
    __shared__ __attribute__((aligned(16))) float    obuf[WAVES][PADF];
    __shared__                              unsigned msk[WAVES][32];

    const int lane = threadIdx.x & 31;
    const int w    = threadIdx.x >> 5;
    const int b    = blockIdx.x * WAVES + w;
    const bool active = (b < B);

    const int seed_dr = p_seed_dr[0];
    const int seed_dc = p_seed_dc[0];
    const int off_r   = p_off_r[0];
    const int off_c   = p_off_c[0];
    const float colf  = (float)p_color[0];

    // LDS byte addresses (low 32 bits of generic pointer == LDS offset)
    const unsigned lds_cells = (unsigned)(uintptr_t)(&cells[w][0]);
    const unsigned lds_obuf  = (unsigned)(uintptr_t)(&obuf[w][0]);

    // ---- async-copy this wave's grid into LDS (225 x B128) ----
    if (active) {
        unsigned long long gbase =
            (unsigned long long)(uintptr_t)(grids + (size_t)b * CELLS);
        for (int k = lane; k < QCELLS; k += 32) {
            unsigned long long ga = gbase + (unsigned)(k * 16);
            unsigned la = lds_cells + (unsigned)(k * 16);
            asm volatile("global_load_async_to_lds_b128 %0, %1, off"
                         :: "v"(la), "v"(ga) : "memory");
        }
        asm volatile("s_wait_asynccnt 0" ::: "memory");
    }

    // ---- flood fill: bitmask fixpoint, one row per lane ----
    int a_r = 0, a_c = 0;
    unsigned m = 0;
    if (active) {
        a_r = anchors[2 * b + 0];
        a_c = anchors[2 * b + 1];
        const int sr = clip30(a_r + seed_dr);
        const int sc = clip30(a_c + seed_dc);
        const float seedcol = cells[w][sr * WID + sc];

        unsigned same = 0;
        if (lane < HGT) {
            for (int c = 0; c < WID; ++c)
                same |= (cells[w][lane * WID + c] == seedcol) ? (1u << c) : 0u;
        }
        m = (lane == sr) ? (1u << sc) : 0u;
        for (;;) {
            unsigned up = __shfl_up(m, 1, 32);
            if (lane == 0) up = 0u;
            unsigned dn = __shfl_down(m, 1, 32);
            unsigned nm = same & (m | (m << 1) | (m >> 1) | up | dn);
            int changed = __any(nm != m);
            m = nm;
            if (!changed) break;
        }
        msk[w][lane] = (lane < HGT) ? m : 0u;
    }
    __syncthreads();

    // ---- centroid of the mask (exact integer sums -> f32, like JAX) ----
    int cnt = __popc(m);
    int sr_sum = lane * cnt;
    int sc_sum = 0;
    {
        unsigned t = m;
        while (t) { sc_sum += __ffs(t) - 1; t &= t - 1; }
    }
    for (int d = 16; d; d >>= 1) {
        cnt    += __shfl_xor(cnt, d, 32);
        sr_sum += __shfl_xor(sr_sum, d, 32);
        sc_sum += __shfl_xor(sc_sum, d, 32);
    }
    const float denom  = fmaxf((float)cnt, 1.0f);
    const float cent_r = (float)sr_sum / denom;
    const float cent_c = (float)sc_sum / denom;
    int ar = 0, ac = 0;
    if (cnt > 0) {
        ar = (int)truncf((float)a_r + (float)off_r - cent_r);
        ac = (int)truncf((float)a_c + (float)off_c - cent_c);
    }

    // ---- composed gather: rotate90 ∘ reflect_x ∘ translate, then paint ----
    if (active && lane < HGT) {
        const int i = lane;
        for (int j = 0; j < WID; ++j) {
            // rotate_90 source
            int ri1 = a_r - (j - a_c);
            int ci1 = a_c + (i - a_r);
            bool v = (ri1 >= 0) & (ri1 < HGT) & (ci1 >= 0) & (ci1 < WID);
            int r1 = clip30(ri1), c1 = clip30(ci1);
            // reflect_x source (only row validity checked, per reference)
            int ri2 = 2 * a_r - r1;
            v = v & (ri2 >= 0) & (ri2 < HGT);
            int r2 = clip30(ri2), c2 = c1;
            // translate source
            int ri3 = r2 - ar;
            int ci3 = c2 - ac;
            v = v & (ri3 >= 0) & (ri3 < HGT) & (ci3 >= 0) & (ci3 < WID);
            int r3 = clip30(ri3), c3 = clip30(ci3);

            unsigned bit = (msk[w][r3] >> c3) & 1u;
            bool sel = v && (bit != 0u);
            obuf[w][i * WID + j] = sel ? colf : cells[w][i * WID + j];
        }
    }
    __syncthreads();

    // ---- async-store the painted tile back (225 x B128) ----
    if (active) {
        unsigned long long obase =
            (unsigned long long)(uintptr_t)(out + (size_t)b * CELLS);
        for (int k = lane; k < QCELLS; k += 32) {
            unsigned long long ga = obase + (unsigned)(k * 16);
            unsigned la = lds_obuf + (unsigned)(k * 16);
            asm volatile("global_store_async_from_lds_b128 %0, %1, off"
                         :: "v"(ga), "v"(la) : "memory");
        }
        asm volatile("s_wait_asynccnt 0" ::: "memory");
    }
}

extern "C" void kernel_launch(void* const* d_in, const int* in_sizes, int n_in,
                              void* d_out, int out_size, void* d_ws, size_t ws_size,
                              hipStream_t stream) {
    const float* grids   = (const float*)d_in[0];
    const int*   anchors = (const int*)d_in[1];
    const int*   seed_dr = (const int*)d_in[2];
    const int*   seed_dc = (const int*)d_in[3];
    const int*   off_r   = (const int*)d_in[4];
    const int*   off_c   = (const int*)d_in[5];
    const int*   color   = (const int*)d_in[6];
    float*       out     = (float*)d_out;

    const int B = in_sizes[0] / CELLS;
    const int blocks = (B + WAVES - 1) / WAVES;
    arc_paint_kernel<<<blocks, 256, 0, stream>>>(
        grids, anchors, seed_dr, seed_dc, off_r, off_c, color, out, B);
}